// TaskAlignedAssigner_27273042329991
// MI455X (gfx1250) — compile-verified
//
#include <hip/hip_runtime.h>
#include <hip/hip_bf16.h>
#include <stdint.h>

#define AS1 __attribute__((address_space(1)))
#define AS3 __attribute__((address_space(3)))

typedef int   v2i __attribute__((ext_vector_type(2)));
typedef float v2f __attribute__((ext_vector_type(2)));
typedef float v8f __attribute__((ext_vector_type(8)));

#define BS   32
#define A    8400
#define C    80
#define N    48
#define TOPK 13
#define TILE 2048
#define EPSF 1e-9f

// ---------- CDNA5 async global -> LDS (ASYNCcnt path) ----------
__device__ __forceinline__ void async_g2l_b64(const void* gsrc, void* ldst) {
#if __has_builtin(__builtin_amdgcn_global_load_async_to_lds_b64)
  // signature: (v2i addrspace(1)*, v2i addrspace(3)*, imm offset, imm cpol)
  __builtin_amdgcn_global_load_async_to_lds_b64((AS1 v2i*)gsrc, (AS3 v2i*)ldst, 0, 0);
#else
  unsigned loff = (unsigned)(unsigned long long)(AS3 char*)ldst;
  asm volatile("global_load_async_to_lds_b64 %0, %1, off"
               :: "v"(loff), "v"((unsigned long long)(size_t)gsrc) : "memory");
#endif
}

__device__ __forceinline__ void wait_async0() {
#if __has_builtin(__builtin_amdgcn_s_wait_asynccnt)
  __builtin_amdgcn_s_wait_asynccnt(0);
#else
  asm volatile("s_wait_asynccnt 0" ::: "memory");
#endif
}

// IoU exactly as reference _cal_iou (box_a = gt, box_b = pred), then clip(0)
__device__ __forceinline__ float iou_gt_pd(float gx1, float gy1, float gx2, float gy2,
                                           const float4 pb) {
  float iw = fminf(gx2, pb.z) - fmaxf(gx1, pb.x); iw = fmaxf(iw, 0.f);
  float ih = fminf(gy2, pb.w) - fmaxf(gy1, pb.y); ih = fmaxf(ih, 0.f);
  float inter = iw * ih;
  float a1 = (gx2 - gx1) * (gy2 - gy1 + EPSF);
  float a2 = (pb.z - pb.x) * (pb.w - pb.y + EPSF);
  float uni = a1 + a2 - inter + EPSF;
  return fmaxf(inter / uni, 0.f);
}

__global__ __launch_bounds__(256) void k_init(unsigned* __restrict__ p, int nwords) {
  int i = blockIdx.x * 256 + threadIdx.x;
  if (i < nwords) p[i] = 0u;
}

// ---------------------------------------------------------------------------
// bbox_scores[b][n][a] = pd_scores[b][a][label[b][n]] as one-hot GEMM:
//   S(48 x 8400) = onehot(labels)(48 x 80) x pd_scores^T(80 x 8400)
// via V_WMMA_F32_16X16X4_F32 (f32, K=4 -> bit-exact: one nonzero product/row).
// One wave per 16-anchor tile; 3 M-tiles (48 gts) x 20 K-steps.
// ---------------------------------------------------------------------------
__global__ __launch_bounds__(256) void k_bbsc(
    const float* __restrict__ pd_scores, const int* __restrict__ gt_labels,
    float* __restrict__ bbsc) {
  const int b    = blockIdx.y;
  const int wave = threadIdx.x >> 5;
  const int lane = threadIdx.x & 31;
  const int tile = blockIdx.x * 8 + wave;
  if (tile >= A / 16) return;  // wave-uniform: EXEC stays all-ones for WMMA
  const int a0 = tile * 16;
  const int lane16 = lane & 15;
  const int hi = lane >> 4;  // 0 -> K pair {0,1}; 1 -> K pair {2,3}

  // A-matrix rows (M = gt index within tile) owned by this lane
  const int lab0 = gt_labels[b * N + 0  + lane16];
  const int lab1 = gt_labels[b * N + 16 + lane16];
  const int lab2 = gt_labels[b * N + 32 + lane16];

  v8f acc0 = {}, acc1 = {}, acc2 = {};
  // B-matrix: B[k][nn] = pd_scores[a0+nn][kbase+k]; lane holds (k=2*hi, 2*hi+1) of col lane16
  const float* prow = pd_scores + ((size_t)b * A + a0 + lane16) * C + 2 * hi;

#pragma unroll
  for (int kk = 0; kk < C / 4; ++kk) {
    const int k0 = kk * 4 + 2 * hi;
    const float2 p = *(const float2*)(prow + kk * 4);
    v2f bm; bm.x = p.x; bm.y = p.y;
    v2f a0m, a1m, a2m;
    a0m.x = (lab0 == k0) ? 1.f : 0.f;  a0m.y = (lab0 == k0 + 1) ? 1.f : 0.f;
    a1m.x = (lab1 == k0) ? 1.f : 0.f;  a1m.y = (lab1 == k0 + 1) ? 1.f : 0.f;
    a2m.x = (lab2 == k0) ? 1.f : 0.f;  a2m.y = (lab2 == k0 + 1) ? 1.f : 0.f;
    acc0 = __builtin_amdgcn_wmma_f32_16x16x4_f32(false, a0m, false, bm, (short)0, acc0, false, false);
    acc1 = __builtin_amdgcn_wmma_f32_16x16x4_f32(false, a1m, false, bm, (short)0, acc1, false, false);
    acc2 = __builtin_amdgcn_wmma_f32_16x16x4_f32(false, a2m, false, bm, (short)0, acc2, false, false);
  }

  // D layout: VGPR r -> row M = r + 8*hi, col = lane16
#pragma unroll
  for (int r = 0; r < 8; ++r) {
    const int m = r + 8 * hi;
    bbsc[((size_t)b * N + m)        * A + a0 + lane16] = acc0[r];
    bbsc[((size_t)b * N + 16 + m)   * A + a0 + lane16] = acc1[r];
    bbsc[((size_t)b * N + 32 + m)   * A + a0 + lane16] = acc2[r];
  }
}

// One block per (b, n): top-13 over A, bit-scatter into mask64, packed argmax_n(iou).
__global__ __launch_bounds__(256) void k_assign(
    const float* __restrict__ pd_scores, const float* __restrict__ pd_bbox,
    const float* __restrict__ anc, const int* __restrict__ gt_labels,
    const float* __restrict__ gt_bbox, const float* __restrict__ mask_gt,
    unsigned long long* __restrict__ mask64, unsigned long long* __restrict__ maxov,
    const float* __restrict__ bbsc, int use_bb) {
  __shared__ float s_anc[TILE * 2];
  __shared__ float s_cval[256 * TOPK];
  __shared__ int   s_cidx[256 * TOPK];
  __shared__ float s_rv[256];
  __shared__ int   s_ri[256];
  __shared__ int   s_rw[256];
  __shared__ int   s_top[TOPK];
  __shared__ float s_gt[4];

  const int bn = blockIdx.x;
  const int b = bn / N, n = bn % N;
  const int tid = threadIdx.x;
  if (tid < 4) s_gt[tid] = gt_bbox[(size_t)(b * N + n) * 4 + tid];
  __syncthreads();
  const float gx1 = s_gt[0], gy1 = s_gt[1], gx2 = s_gt[2], gy2 = s_gt[3];
  const int   label = gt_labels[b * N + n];
  const float mg    = mask_gt[b * N + n];
  const float* bbrow = bbsc + (size_t)(b * N + n) * A;

  float tv[TOPK]; int ti[TOPK];
#pragma unroll
  for (int k = 0; k < TOPK; k++) { tv[k] = -1.f; ti[k] = 0x7FFFFFFF; }

  const float4* pb4 = (const float4*)pd_bbox;

  for (int t0 = 0; t0 < A; t0 += TILE) {
    const int tc = (A - t0 < TILE) ? (A - t0) : TILE;
    // stage anchor points for this tile via async DMA to LDS
    for (int i = tid; i < tc; i += 256)
      async_g2l_b64(anc + (size_t)(t0 + i) * 2, &s_anc[i * 2]);
    wait_async0();
    __syncthreads();

    for (int i = tid; i < tc; i += 256) {
      const int a = t0 + i;
      const float4 pb = pb4[(size_t)b * A + a];
      const float o = iou_gt_pd(gx1, gy1, gx2, gy2, pb);
      // packed argmax over n of overlaps (tie -> lowest n), order-independent
      unsigned long long key =
          ((unsigned long long)__float_as_uint(o) << 32) |
          (unsigned long long)(0xFFFFFFFFu - (unsigned)n);
      atomicMax(&maxov[(size_t)b * A + a], key);

      const int ap = (a + 256 < A) ? (a + 256) : a;
      float sc;
      if (use_bb) {
        __builtin_prefetch(bbrow + ap, 0, 1);
        sc = bbrow[a];
      } else {
        __builtin_prefetch(pd_scores + ((size_t)b * A + ap) * C + label, 0, 1);
        sc = pd_scores[((size_t)b * A + a) * C + label];
      }
      const float o2 = o * o;
      const float al = sc * (o2 * o2 * o2);  // score^1 * iou^6
      const float ax = s_anc[i * 2], ay = s_anc[i * 2 + 1];
      const float dmin = fminf(fminf(ax - gx1, ay - gy1), fminf(gx2 - ax, gy2 - ay));
      const float m = (dmin > EPSF) ? al : 0.f;

      const bool better = (m > tv[TOPK - 1]) || (m == tv[TOPK - 1] && a < ti[TOPK - 1]);
      if (better) {
        tv[TOPK - 1] = m; ti[TOPK - 1] = a;
#pragma unroll
        for (int k = TOPK - 1; k > 0; --k) {
          bool sw = (tv[k] > tv[k - 1]) || (tv[k] == tv[k - 1] && ti[k] < ti[k - 1]);
          float va = sw ? tv[k] : tv[k - 1];
          float vb = sw ? tv[k - 1] : tv[k];
          int ia = sw ? ti[k] : ti[k - 1];
          int ib = sw ? ti[k - 1] : ti[k];
          tv[k - 1] = va; tv[k] = vb; ti[k - 1] = ia; ti[k] = ib;
        }
      }
    }
    __syncthreads();  // done reading s_anc before next tile restages
  }

#pragma unroll
  for (int k = 0; k < TOPK; k++) {
    s_cval[tid * TOPK + k] = tv[k];
    s_cidx[tid * TOPK + k] = ti[k];
  }
  __syncthreads();

  // 13 rounds of block-wide argmax (tie -> lowest anchor index) == lax.top_k order
  for (int r = 0; r < TOPK; r++) {
    float bv = -3.f; int bi = 0x7FFFFFFF; int bslot = 0;
#pragma unroll
    for (int k = 0; k < TOPK; k++) {
      float v = s_cval[tid * TOPK + k]; int id = s_cidx[tid * TOPK + k];
      if (v > bv || (v == bv && id < bi)) { bv = v; bi = id; bslot = k; }
    }
    s_rv[tid] = bv; s_ri[tid] = bi; s_rw[tid] = tid;
    __syncthreads();
    for (int s = 128; s > 0; s >>= 1) {
      if (tid < s) {
        float v2 = s_rv[tid + s]; int i2 = s_ri[tid + s];
        if (v2 > s_rv[tid] || (v2 == s_rv[tid] && i2 < s_ri[tid])) {
          s_rv[tid] = v2; s_ri[tid] = i2; s_rw[tid] = s_rw[tid + s];
        }
      }
      __syncthreads();
    }
    if (tid == 0) s_top[r] = s_ri[0];
    if (tid == s_rw[0]) { s_cval[tid * TOPK + bslot] = -2.f; s_cidx[tid * TOPK + bslot] = 0x7FFFFFFF; }
    __syncthreads();
  }

  // is_in_topk * mask_in_gts * mask_gt -> set bit n on winning anchors.
  // (mask_gt<=0 reproduces the reference's duplicate-at-0 zeroing: no bits.)
  if (tid < TOPK && mg > 0.f) {
    const int a = s_top[tid];
    if (a >= 0 && a < A) {
      const float ax = anc[2 * a], ay = anc[2 * a + 1];
      const float dmin = fminf(fminf(ax - gx1, ay - gy1), fminf(gx2 - ax, gy2 - ay));
      if (dmin > EPSF) atomicOr(&mask64[(size_t)b * A + a], 1ull << n);
    }
  }
}

// Per (b,a): resolve multi-assignment, target_gt_idx/fg, fold pos_align/pos_overlap.
__global__ __launch_bounds__(256) void k_resolve(
    const float* __restrict__ pd_scores, const float* __restrict__ pd_bbox,
    const int* __restrict__ gt_labels, const float* __restrict__ gt_bbox,
    const unsigned long long* __restrict__ mask64,
    const unsigned long long* __restrict__ maxov,
    int* __restrict__ tgi, int* __restrict__ fg, float* __restrict__ alsel,
    unsigned* __restrict__ posal, unsigned* __restrict__ posov,
    const float* __restrict__ bbsc, int use_bb) {
  const int idx = blockIdx.x * 256 + threadIdx.x;
  if (idx >= BS * A) return;
  const int b = idx / A;
  const int a = idx % A;
  unsigned long long m = mask64[idx];
  if (__popcll(m) > 1) {
    const unsigned long long key = maxov[idx];
    const unsigned nn = 0xFFFFFFFFu - (unsigned)(key & 0xFFFFFFFFull);
    m = (nn < (unsigned)N) ? (1ull << nn) : 0ull;
  }
  int t = 0, f = 0;
  if (m) { t = __ffsll(m) - 1; f = 1; }
  tgi[idx] = t; fg[idx] = f;
  float al = 0.f;
  if (f) {
    const float4 gb = ((const float4*)gt_bbox)[b * N + t];
    const float4 pb = ((const float4*)pd_bbox)[idx];
    const float o = iou_gt_pd(gb.x, gb.y, gb.z, gb.w, pb);
    float sc;
    if (use_bb) {
      sc = bbsc[(size_t)(b * N + t) * A + a];
    } else {
      const int lbl = gt_labels[b * N + t];
      sc = pd_scores[(size_t)idx * C + lbl];
    }
    const float o2 = o * o;
    al = sc * (o2 * o2 * o2);
    atomicMax(&posal[b * N + t], __float_as_uint(al));  // floats >= 0: bit-monotone
    atomicMax(&posov[b * N + t], __float_as_uint(o));
  }
  alsel[idx] = al;
}

// Per (b,a): final outputs. Output order: labels | bbox | scores | fg | gt_idx (all f32).
__global__ __launch_bounds__(256) void k_write(
    const int* __restrict__ gt_labels, const float* __restrict__ gt_bbox,
    const int* __restrict__ tgi, const int* __restrict__ fg,
    const float* __restrict__ alsel,
    const unsigned* __restrict__ posal, const unsigned* __restrict__ posov,
    float* __restrict__ out) {
  const int idx = blockIdx.x * 256 + threadIdx.x;
  if (idx >= BS * A) return;
  const int b = idx / A;
  const int t = tgi[idx];
  const int f = fg[idx];
  const int lbl = gt_labels[b * N + t];
  const float4 gb = ((const float4*)gt_bbox)[b * N + t];
  float norm = 0.f;
  if (f) {
    const float pa = __uint_as_float(posal[b * N + t]);
    const float po = __uint_as_float(posov[b * N + t]);
    norm = alsel[idx] * po / (pa + EPSF);  // mask_pos has <=1 bit -> max over n collapses
  }
  float* out_lab = out;
  float* out_bb  = out + (size_t)BS * A;
  float* out_sc  = out + (size_t)BS * A * 5;
  float* out_fg  = out + (size_t)BS * A * 85;
  float* out_tg  = out + (size_t)BS * A * 86;

  out_lab[idx] = (float)lbl;
  ((float4*)out_bb)[idx] = gb;
  out_fg[idx] = f ? 1.f : 0.f;
  out_tg[idx] = (float)t;

  float4* sp = (float4*)(out_sc + (size_t)idx * C);
  const float4 z = make_float4(0.f, 0.f, 0.f, 0.f);
#pragma unroll
  for (int q = 0; q < C / 4; q++) sp[q] = z;  // b128 zero-fill (86 MB stream)
  if (f) out_sc[(size_t)idx * C + lbl] = norm;
}

extern "C" void kernel_launch(void* const* d_in, const int* in_sizes, int n_in,
                              void* d_out, int out_size, void* d_ws, size_t ws_size,
                              hipStream_t stream) {
  const float* pd_scores = (const float*)d_in[0];
  const float* pd_bbox   = (const float*)d_in[1];
  const float* anc       = (const float*)d_in[2];
  const int*   gt_labels = (const int*)d_in[3];
  const float* gt_bbox   = (const float*)d_in[4];
  const float* mask_gt   = (const float*)d_in[5];
  float* out = (float*)d_out;
  char* ws = (char*)d_ws;

  // workspace layout (base 7.54 MB + optional 51.6 MB bbox_scores)
  unsigned long long* mask64 = (unsigned long long*)(ws);            // 2,150,400 B
  unsigned long long* maxov  = (unsigned long long*)(ws + 2150400);  // 2,150,400 B
  unsigned* posal = (unsigned*)(ws + 4300800);                       //     6,144 B
  unsigned* posov = (unsigned*)(ws + 4306944);                       //     6,144 B
  int*   tgi   = (int*)(ws + 4313088);                               // 1,075,200 B
  int*   fg    = (int*)(ws + 5388288);                               // 1,075,200 B
  float* alsel = (float*)(ws + 6463488);                             // 1,075,200 B
  const size_t WS_BASE = 7538688;
  const size_t BB_BYTES = (size_t)BS * N * A * sizeof(float);        // 51,609,600 B
  float* bbsc = (float*)(ws + WS_BASE);
  const int use_bb = (ws_size >= WS_BASE + BB_BYTES) ? 1 : 0;

  const int zwords = 4313088 / 4;  // zero mask64 + maxov + posal + posov
  k_init<<<(zwords + 255) / 256, 256, 0, stream>>>((unsigned*)ws, zwords);
  if (use_bb) {
    // 525 anchor tiles of 16, 8 waves/block -> 66 blocks per batch image
    k_bbsc<<<dim3(66, BS), 256, 0, stream>>>(pd_scores, gt_labels, bbsc);
  }
  k_assign<<<BS * N, 256, 0, stream>>>(pd_scores, pd_bbox, anc, gt_labels, gt_bbox,
                                       mask_gt, mask64, maxov, bbsc, use_bb);
  k_resolve<<<(BS * A + 255) / 256, 256, 0, stream>>>(pd_scores, pd_bbox, gt_labels,
                                                      gt_bbox, mask64, maxov, tgi, fg,
                                                      alsel, posal, posov, bbsc, use_bb);
  k_write<<<(BS * A + 255) / 256, 256, 0, stream>>>(gt_labels, gt_bbox, tgi, fg, alsel,
                                                    posal, posov, out);
}